// DynamicMemory_18468359373437
// MI455X (gfx1250) — compile-verified
//
#include <hip/hip_runtime.h>
#include <hip/hip_bf16.h>

// DynamicMemory routing on MI455X (gfx1250, wave32).
//
// Phase 1 (dominant, memory-bound): u[b,i,t,d] = sum_k W[i,t,d,k]*cat[b,t,k]
//   W = 335.5 MB streamed once -> ~14.4 us floor @ 23.3 TB/s. f32 WMMA
//   (v_wmma_f32_16x16x4_f32) gives exact math; compute is free vs. the stream.
// Phase 2: 3 routing iterations over u (41.9 MB, L2-resident).

typedef __attribute__((ext_vector_type(2))) float v2f;
typedef __attribute__((ext_vector_type(4))) float v4f;
typedef __attribute__((ext_vector_type(8))) float v8f;

#define B_  8
#define M_  64
#define T_  256
#define MT  320           // M_ + T_
#define D_  64
#define TB  8             // t's per workgroup in kernel 1
#define RS  (TB * 64 + 2) // LDS row stride (floats) per padded-b row: breaks bank conflicts

// ---------------------------------------------------------------------------
// Kernel 1: u[b,i,t,d] = sum_k W[i,t,d,k] * cat[b,t,k], WMMA f32 16x16x4.
// Grid: (40 t-blocks, 64 i). Block: 128 threads = 4 waves; wave w owns d-tile
// [16w,16w+16). N dimension = b (0..7 live, 8..15 zero-padded in LDS).
// ---------------------------------------------------------------------------
__global__ void __launch_bounds__(128)
dm_u_wmma_kernel(const float* __restrict__ W,
                 const float* __restrict__ M_emb,
                 const float* __restrict__ Ht_n,
                 float* __restrict__ u_ws)
{
    const int tb  = blockIdx.x;   // 0..39
    const int i   = blockIdx.y;   // 0..63
    const int tid = threadIdx.x;  // 0..127

    __shared__ float lds_cat[16 * RS];  // 32,896 B: cat[b(16 padded)][tl*64+k]

    // Zero the whole tile (provides the b=8..15 zero columns).
    for (int idx = tid; idx < 16 * RS; idx += 128) lds_cat[idx] = 0.0f;
    __syncthreads();

    // Stage cat[b, t, k] for b<8, t in this t-block. cat = concat(M_emb, Ht_n).
    for (int idx = tid; idx < B_ * TB * 64; idx += 128) {
        const int bb  = idx >> 9;          // idx / 512
        const int rem = idx & 511;
        const int tl  = rem >> 6;
        const int k   = rem & 63;
        const int t   = tb * TB + tl;
        const float v = (t < M_) ? M_emb[((size_t)bb * M_ + t) * D_ + k]
                                 : Ht_n[((size_t)bb * T_ + (t - M_)) * D_ + k];
        lds_cat[bb * RS + tl * 64 + k] = v;
    }
    __syncthreads();

    const int w       = tid >> 5;        // wave id 0..3 -> d-tile
    const int lane    = tid & 31;
    const int lan16   = lane & 15;       // A: M row within tile / B: N (=b)
    const int halfsel = lane >> 4;       // upper half-wave handles K+2 / M+8

    const float* Wi = W + (size_t)i * MT * (D_ * D_);   // W[i,:,:,:]

    for (int tl = 0; tl < TB; ++tl) {
        const int t = tb * TB + tl;
        // A fragment source: W[i, t, d_row, k0 + {0,1}], d_row = 16w + lan16
        const float* Wt = Wi + (size_t)t * (D_ * D_) + (w * 16 + lan16) * D_ + 2 * halfsel;
        // B fragment source: lds_cat[b=lan16][tl*64 + k0 + {0,1}]
        const float* Bt = lds_cat + lan16 * RS + tl * 64 + 2 * halfsel;

        // Prefetch next t's W tile for this wave while we compute.
        if (tl + 1 < TB) __builtin_prefetch(Wt + D_ * D_, 0, 1);

        v8f acc = {};
        #pragma unroll
        for (int kb = 0; kb < 16; ++kb) {
            v2f a = *(const v2f*)(Wt + 4 * kb);   // global_load_b64 (8B aligned)
            v2f b = *(const v2f*)(Bt + 4 * kb);   // ds_load_b64     (8B aligned)
            // (neg_a, A, neg_b, B, c_mod, C, reuse_a, reuse_b)
            acc = __builtin_amdgcn_wmma_f32_16x16x4_f32(
                      false, a, false, b, (short)0, acc, false, false);
        }

        // C/D layout: VGPR r holds (d = 16w + r + 8*halfsel, b = lan16).
        const int bb = lan16;
        if (bb < B_) {
            const size_t off = (((size_t)(bb * M_ + i) * MT + t) * D_)
                               + w * 16 + 8 * halfsel;  // 8-float contiguous run
            v4f lo = {acc[0], acc[1], acc[2], acc[3]};
            v4f hi = {acc[4], acc[5], acc[6], acc[7]};
            *(v4f*)(u_ws + off)     = lo;   // 32B-aligned
            *(v4f*)(u_ws + off + 4) = hi;   // 16B-aligned
        }
    }
}

// ---------------------------------------------------------------------------
// Kernel 2: 3 routing iterations per (b,i). One block per (b,i); u slice
// (80 KB) is L2-resident. Grid 512, block 256.
// ---------------------------------------------------------------------------
__global__ void __launch_bounds__(256)
dm_iter_kernel(const float* __restrict__ u_ws,
               const float* __restrict__ m_init,
               float* __restrict__ out)
{
    const int bi  = blockIdx.x;    // b*64 + i
    const int tid = threadIdx.x;   // 0..255
    const float* u = u_ws + (size_t)bi * MT * D_;

    __shared__ float sm_m[D_];
    __shared__ float sm_e[MT];     // logits, then exp(logit - max)
    __shared__ float sm_s[D_];
    __shared__ float sm_red[64];
    __shared__ float sm_scalar;

    if (tid < D_) sm_m[tid] = m_init[(size_t)bi * D_ + tid];
    __syncthreads();

    for (int it = 0; it < 3; ++it) {
        // logits[t] = sum_d m[d] * u[t,d]
        for (int t = tid; t < MT; t += 256) {
            const float* ur = u + (size_t)t * D_;
            float acc = 0.0f;
            #pragma unroll 8
            for (int d = 0; d < D_; ++d) acc += sm_m[d] * ur[d];
            sm_e[t] = acc;
        }
        __syncthreads();

        // softmax max-reduce
        if (tid < 64) {
            float mx = -3.402823466e38f;
            for (int t = tid; t < MT; t += 64) mx = fmaxf(mx, sm_e[t]);
            sm_red[tid] = mx;
        }
        __syncthreads();
        if (tid == 0) {
            float mx = sm_red[0];
            for (int j = 1; j < 64; ++j) mx = fmaxf(mx, sm_red[j]);
            sm_scalar = mx;
        }
        __syncthreads();
        const float gmax = sm_scalar;
        __syncthreads();

        // exp + sum
        if (tid < 64) {
            float ssum = 0.0f;
            for (int t = tid; t < MT; t += 64) {
                const float e = expf(sm_e[t] - gmax);
                sm_e[t] = e;
                ssum += e;
            }
            sm_red[tid] = ssum;
        }
        __syncthreads();
        if (tid == 0) {
            float s = 0.0f;
            for (int j = 0; j < 64; ++j) s += sm_red[j];
            sm_scalar = 1.0f / s;
        }
        __syncthreads();
        const float inv = sm_scalar;
        __syncthreads();

        // s[d] = tanh(sum_t alph[t] * u[t,d]);  alph = e * inv
        if (tid < D_) {
            float acc = 0.0f;
            for (int t = 0; t < MT; ++t) acc += sm_e[t] * u[(size_t)t * D_ + tid];
            const float sv = tanhf(acc * inv);
            sm_s[tid]   = sv;
            sm_red[tid] = sv * sv;
        }
        __syncthreads();
        if (tid == 0) {
            float q = 0.0f;
            for (int j = 0; j < 64; ++j) q += sm_red[j];
            const float n = sqrtf(q) + 1e-4f;
            sm_scalar = n / (1.0f + n * n);   // (n^2/(1+n^2)) * (1/n)
        }
        __syncthreads();
        if (tid < D_) sm_m[tid] = sm_scalar * sm_s[tid];
        __syncthreads();
    }

    if (tid < D_) out[(size_t)bi * D_ + tid] = sm_m[tid];
}

// ---------------------------------------------------------------------------
extern "C" void kernel_launch(void* const* d_in, const int* in_sizes, int n_in,
                              void* d_out, int out_size, void* d_ws, size_t ws_size,
                              hipStream_t stream)
{
    (void)in_sizes; (void)n_in; (void)out_size; (void)ws_size;
    const float* M_emb  = (const float*)d_in[0];  // [8,64,64]
    const float* Ht_n   = (const float*)d_in[1];  // [8,256,64]
    const float* m_init = (const float*)d_in[2];  // [8,64,64]
    const float* W      = (const float*)d_in[3];  // [64,320,64,64]
    float* u_ws = (float*)d_ws;                   // needs 8*64*320*64*4 = 41,943,040 B

    dim3 g1(MT / TB, M_);  // 40 x 64 blocks
    dm_u_wmma_kernel<<<g1, 128, 0, stream>>>(W, M_emb, Ht_n, u_ws);
    dm_iter_kernel<<<B_ * M_, 256, 0, stream>>>(u_ws, m_init, (float*)d_out);
}